// GCN_19344532701547
// MI455X (gfx1250) — compile-verified
//
#include <hip/hip_runtime.h>

typedef float v2f __attribute__((ext_vector_type(2)));
typedef float v8f __attribute__((ext_vector_type(8)));

// ---------------------------------------------------------------------------
// Degree / normalization kernels (computed once, reused by all 3 layers)
// ---------------------------------------------------------------------------
__global__ void deg_init_kernel(float* __restrict__ deg, int n) {
    int i = blockIdx.x * blockDim.x + threadIdx.x;
    if (i < n) deg[i] = 1.0f;  // self loop contributes 1 to every node degree
}

__global__ void deg_count_kernel(const int* __restrict__ dst, float* __restrict__ deg, int e) {
    int i = blockIdx.x * blockDim.x + threadIdx.x;
    if (i < e) atomicAdd(&deg[dst[i]], 1.0f);
}

__global__ void dinv_kernel(float* __restrict__ deg, int n) {
    int i = blockIdx.x * blockDim.x + threadIdx.x;
    if (i < n) deg[i] = rsqrtf(deg[i]);  // deg >= 1 always (self loop)
}

// ---------------------------------------------------------------------------
// Dense transform: hs = dinv .* (x @ W), acc initialized to hs (self loop).
// One wave32 owns a 16-row tile; V_WMMA_F32_16X16X4_F32 accumulates K in
// steps of 4; COUT/16 column tiles. fp32 end-to-end (matches reference).
// All fragment loads are branch-free (address clamping / cndmask selects).
// ---------------------------------------------------------------------------
template <int CIN, int COUT>
__global__ __launch_bounds__(256) void transform_kernel(
    const float* __restrict__ x, const float* __restrict__ W,
    const float* __restrict__ dinv, float* __restrict__ hs,
    float* __restrict__ acc_out, int n)
{
    constexpr int  KSTEPS = (CIN + 3) / 4;
    constexpr int  TILES  = COUT / 16;
    constexpr bool KPAD   = (CIN % 4) != 0;   // only layer 1 (CIN = 11)

    int wave     = (blockIdx.x * blockDim.x + threadIdx.x) >> 5;
    int lane     = threadIdx.x & 31;
    int row_base = wave * 16;
    if (row_base >= n) return;                // wave-uniform: EXEC all-1s below

    int m  = lane & 15;                      // A row / B,D column within tile
    int kh = (lane >> 4) & 1;                // lane half: K pair / D row half
    int arow = row_base + m;
    // Clamp instead of guard: an OOB A-row only feeds an OOB D-row, which is
    // never stored, so garbage values there are harmless.
    int ar = arow < n ? arow : n - 1;

    v8f acc[TILES];
#pragma unroll
    for (int t = 0; t < TILES; ++t) acc[t] = {};

#pragma unroll
    for (int kb = 0; kb < KSTEPS; ++kb) {
        int k0 = kb * 4 + 2 * kh;
        // ---- A fragment: lane holds x[ar][k0], x[ar][k0+1] ----
        v2f a;
        if constexpr (!KPAD) {
            // CIN even & k0 even -> 8B-aligned pair load
            a = *(const v2f*)&x[ar * CIN + k0];
        } else {
            // clamp addresses; padded positions are zeroed via B instead
            int ka0 = (k0     < CIN) ? k0     : CIN - 1;
            int ka1 = (k0 + 1 < CIN) ? k0 + 1 : CIN - 1;
            a.x = x[ar * CIN + ka0];
            a.y = x[ar * CIN + ka1];
        }
#pragma unroll
        for (int t = 0; t < TILES; ++t) {
            int ncol = t * 16 + m;
            // ---- B fragment: lane holds W[k0][ncol], W[k0+1][ncol] ----
            v2f b;
            if constexpr (!KPAD) {
                b.x = W[ k0      * COUT + ncol];
                b.y = W[(k0 + 1) * COUT + ncol];
            } else {
                bool ok0 = (k0     < CIN);
                bool ok1 = (k0 + 1 < CIN);
                float w0 = W[(ok0 ? k0     : 0) * COUT + ncol];
                float w1 = W[(ok1 ? k0 + 1 : 0) * COUT + ncol];
                b.x = ok0 ? w0 : 0.0f;    // zero padded K rows of B
                b.y = ok1 ? w1 : 0.0f;
            }
            acc[t] = __builtin_amdgcn_wmma_f32_16x16x4_f32(
                /*neg_a=*/false, a, /*neg_b=*/false, b,
                /*c_mod=*/(short)0, acc[t], /*reuse_a=*/false, /*reuse_b=*/false);
        }
    }

    // D layout: VGPR v holds row (row_base + v + 8*kh), column (t*16 + m)
    if (row_base + 16 <= n) {                // wave-uniform fast path
#pragma unroll
        for (int v = 0; v < 8; ++v) {
            int r = row_base + v + 8 * kh;
            float di = dinv[r];
#pragma unroll
            for (int t = 0; t < TILES; ++t) {
                float val = acc[t][v] * di;
                int idx = r * COUT + t * 16 + m;
                hs[idx]      = val;          // gather source for scatter pass
                acc_out[idx] = val;          // accumulator init = self-loop term
            }
        }
    } else {                                 // partial last tile
#pragma unroll
        for (int v = 0; v < 8; ++v) {
            int r = row_base + v + 8 * kh;
            if (r < n) {
                float di = dinv[r];
#pragma unroll
                for (int t = 0; t < TILES; ++t) {
                    float val = acc[t][v] * di;
                    int idx = r * COUT + t * 16 + m;
                    hs[idx]      = val;
                    acc_out[idx] = val;
                }
            }
        }
    }
}

// ---------------------------------------------------------------------------
// Edge scatter: acc[dst] += hs[src].  One thread per (edge, 4 channels):
// 16B-aligned b128 gather + 4 consecutive no-return global_atomic_add_f32.
// Node tables are L2-resident (<= 25.6 MB each vs 192 MB L2).
// ---------------------------------------------------------------------------
template <int C>
__global__ __launch_bounds__(256) void scatter_kernel(
    const int* __restrict__ src, const int* __restrict__ dst,
    const float* __restrict__ hs, float* __restrict__ acc, int e)
{
    constexpr int G = C / 4;                 // thread-groups per edge (pow2)
    int idx = blockIdx.x * blockDim.x + threadIdx.x;
    int ei  = idx / G;
    if (ei >= e) return;
    int c4 = (idx & (G - 1)) * 4;
    int s = src[ei];
    int d = dst[ei];
    const float4 v = *(const float4*)&hs[s * C + c4];
    float* a = &acc[d * C + c4];
    atomicAdd(a + 0, v.x);
    atomicAdd(a + 1, v.y);
    atomicAdd(a + 2, v.z);
    atomicAdd(a + 3, v.w);
}

// ---------------------------------------------------------------------------
// Finalize: out = [relu]( dinv[node] * acc + bias ), float4 per thread
// ---------------------------------------------------------------------------
template <int C, bool RELU>
__global__ __launch_bounds__(256) void finalize_kernel(
    const float* __restrict__ acc, const float* __restrict__ dinv,
    const float* __restrict__ bias, float* __restrict__ out, int n)
{
    constexpr int G = C / 4;
    int idx = blockIdx.x * blockDim.x + threadIdx.x;
    if (idx >= n * G) return;
    int node = idx / G;
    int c4   = (idx & (G - 1)) * 4;
    float di = dinv[node];
    float4 v = *(const float4*)&acc[node * C + c4];
    float4 b = *(const float4*)&bias[c4];
    float4 o;
    o.x = fmaf(di, v.x, b.x);
    o.y = fmaf(di, v.y, b.y);
    o.z = fmaf(di, v.z, b.z);
    o.w = fmaf(di, v.w, b.w);
    if (RELU) {
        o.x = fmaxf(o.x, 0.0f); o.y = fmaxf(o.y, 0.0f);
        o.z = fmaxf(o.z, 0.0f); o.w = fmaxf(o.w, 0.0f);
    }
    *(float4*)&out[node * C + c4] = o;
}

// ---------------------------------------------------------------------------
extern "C" void kernel_launch(void* const* d_in, const int* in_sizes, int n_in,
                              void* d_out, int out_size, void* d_ws, size_t ws_size,
                              hipStream_t stream) {
    const float* x  = (const float*)d_in[0];
    const int*   ei = (const int*)  d_in[1];
    const float* W1 = (const float*)d_in[2];
    const float* b1 = (const float*)d_in[3];
    const float* W2 = (const float*)d_in[4];
    const float* b2 = (const float*)d_in[5];
    const float* W3 = (const float*)d_in[6];
    const float* b3 = (const float*)d_in[7];
    float* out = (float*)d_out;

    const int n = in_sizes[0] / 11;
    const int e = in_sizes[1] / 2;
    const int* src = ei;
    const int* dst = ei + e;

    // Workspace layout (floats): dinv[n] | hs[64n] | acc[64n] | hbuf[32n]
    float* ws   = (float*)d_ws;
    float* dinv = ws;
    float* hs   = dinv + n;
    float* acc  = hs + (size_t)n * 64;
    float* hbuf = acc + (size_t)n * 64;

    const int B = 256;
    auto cdiv = [](long a, long b) { return (int)((a + b - 1) / b); };

    // --- normalization (once) ---
    deg_init_kernel <<<cdiv(n, B), B, 0, stream>>>(dinv, n);
    deg_count_kernel<<<cdiv(e, B), B, 0, stream>>>(dst, dinv, e);
    dinv_kernel     <<<cdiv(n, B), B, 0, stream>>>(dinv, n);

    const int tb = cdiv((long)((n + 15) / 16) * 32, B);  // blocks for transform

    // --- layer 1: 11 -> 16, relu ---
    transform_kernel<11, 16><<<tb, B, 0, stream>>>(x, W1, dinv, hs, acc, n);
    scatter_kernel<16>       <<<cdiv((long)e * 4,  B), B, 0, stream>>>(src, dst, hs, acc, e);
    finalize_kernel<16, true><<<cdiv((long)n * 4,  B), B, 0, stream>>>(acc, dinv, b1, hbuf, n);

    // --- layer 2: 16 -> 32, relu ---
    transform_kernel<16, 32><<<tb, B, 0, stream>>>(hbuf, W2, dinv, hs, acc, n);
    scatter_kernel<32>       <<<cdiv((long)e * 8,  B), B, 0, stream>>>(src, dst, hs, acc, e);
    finalize_kernel<32, true><<<cdiv((long)n * 8,  B), B, 0, stream>>>(acc, dinv, b2, hbuf, n);

    // --- layer 3: 32 -> 64, no relu ---
    transform_kernel<32, 64><<<tb, B, 0, stream>>>(hbuf, W3, dinv, hs, acc, n);
    scatter_kernel<64>        <<<cdiv((long)e * 16, B), B, 0, stream>>>(src, dst, hs, acc, e);
    finalize_kernel<64, false><<<cdiv((long)n * 16, B), B, 0, stream>>>(acc, dinv, b3, out, n);
}